// Spell_7499012899510
// MI455X (gfx1250) — compile-verified
//
#include <hip/hip_runtime.h>
#include <hip/hip_bf16.h>
#include <math.h>

// ---- problem constants (fixed by the reference) ----
#define B_ 16
#define L_ 2048
#define T_ 256
#define D_ 256
#define H_ 512
#define V_ 34

typedef __attribute__((ext_vector_type(2))) float v2f;
typedef __attribute__((ext_vector_type(8))) float v8f;

__device__ __forceinline__ float sigf(float x) { return 1.0f / (1.0f + expf(-x)); }

#define WMMA_F32(a, b, c) \
  __builtin_amdgcn_wmma_f32_16x16x4_f32(false, (a), false, (b), (short)0, (c), false, false)

// ---- workspace layout (floats) ----
enum : size_t {
  OFF_H1   = 0,                       // 2 * 8192 (ping-pong [16,512])
  OFF_C1   = OFF_H1 + 2 * 8192,       // 8192
  OFF_H2   = OFF_C1 + 8192,           // 2 * 4096 (ping-pong [16,256])
  OFF_C2   = OFF_H2 + 2 * 4096,       // 4096
  OFF_CTX  = OFF_C2 + 4096,           // 4096
  OFF_E    = OFF_CTX + 4096,          // 16*2048
  OFF_SUM  = OFF_E + 16 * 2048,       // 32*16
  OFF_CTXP = OFF_SUM + 32 * 16,       // 32*16*256
  WS_FLOATS = OFF_CTXP + 32 * 16 * 256
};

// ---------------- init: copy initial state ----------------
__global__ __launch_bounds__(256) void k_init(const float* __restrict__ h10,
                                              const float* __restrict__ c10,
                                              const float* __restrict__ h20,
                                              const float* __restrict__ c20,
                                              float* __restrict__ ws) {
  int i = blockIdx.x * 256 + threadIdx.x;
  if (i < B_ * H_) { ws[OFF_H1 + i] = h10[i]; ws[OFF_C1 + i] = c10[i]; }
  if (i < B_ * D_) {
    ws[OFF_H2 + i]  = h20[i];
    ws[OFF_C2 + i]  = c20[i];
    ws[OFF_CTX + i] = 0.0f;
  }
}

// straight-line K-segment: no per-iteration region branching.
// A-frag: lanes 0-15 hold K+0,K+1 of row m; lanes 16-31 hold K+2,K+3.
// B-frag: lanes 0-15 hold K+0,K+1 of col n; lanes 16-31 hold K+2,K+3.
__device__ __forceinline__ void gemm_seg(const float* __restrict__ A,
                                         const float* __restrict__ Bp0,
                                         const float* __restrict__ Bp1,
                                         int len, v8f& acc0, v8f& acc1) {
#pragma unroll 4
  for (int k = 0; k < len; k += 4) {
    v2f a  = *(const v2f*)(A + k);
    v2f b0 = *(const v2f*)(Bp0 + k);
    v2f b1 = *(const v2f*)(Bp1 + k);
    acc0 = WMMA_F32(a, b0, acc0);
    acc1 = WMMA_F32(a, b1, acc1);
  }
}

// ---------------- LSTM cell 1 (M=16, N=2048, K=1024) ----------------
// grid = H/64 = 8 blocks, 256 threads (8 waves). Each block computes the
// 64-wide column slice of all four gates -> fused sigmoid/tanh epilogue.
__global__ __launch_bounds__(256) void k_lstm1(
    const float* __restrict__ emb, const int* __restrict__ tr,
    const float* __restrict__ Wih, const float* __restrict__ Whh,
    const float* __restrict__ bih, const float* __restrict__ bhh,
    const float* __restrict__ ctx, const float* __restrict__ h1cur,
    float* __restrict__ h1nxt, float* __restrict__ c1, int t) {
  __shared__ float gLDS[B_ * 256];
  const int tid = threadIdx.x;
  const int wave = tid >> 5, lane = tid & 31;
  const int m = lane & 15, hi2 = (lane >> 4) * 2;
  const int j0 = blockIdx.x * 64;

  // A rows: x = [emb[tok] | ctx | h1_prev], K = 1024 (lane-offset baked in)
  const int tok = tr[m * T_ + t];
  const float* arow_e = emb   + (size_t)tok * D_ + hi2;
  const float* arow_c = ctx   + (size_t)m * D_ + hi2;
  const float* arow_h = h1cur + (size_t)m * H_ + hi2;

  const int tn0 = wave * 2, tn1 = tn0 + 1;
  const int gate = tn0 >> 2;                       // both tiles: same gate
  const int n0 = gate * H_ + j0 + (tn0 & 3) * 16 + m;
  const int n1 = gate * H_ + j0 + (tn1 & 3) * 16 + m;
  const float* b0i = Wih + (size_t)n0 * 512 + hi2;
  const float* b1i = Wih + (size_t)n1 * 512 + hi2;
  const float* b0h = Whh + (size_t)n0 * 512 + hi2;
  const float* b1h = Whh + (size_t)n1 * 512 + hi2;

  v8f acc0 = {}, acc1 = {};
  gemm_seg(arow_e, b0i,       b1i,       256, acc0, acc1);  // k in [0,256)
  gemm_seg(arow_c, b0i + 256, b1i + 256, 256, acc0, acc1);  // k in [256,512)
  gemm_seg(arow_h, b0h,       b1h,       512, acc0, acc1);  // k in [512,1024)

  // C/D layout: VGPR r, lanes<16 -> M=r; lanes>=16 -> M=r+8; N=lane&15
  const int hi8 = (hi2 >> 1) * 8;
  const int col0 = gate * 64 + (tn0 & 3) * 16 + m;
  const int col1 = gate * 64 + (tn1 & 3) * 16 + m;
#pragma unroll
  for (int r = 0; r < 8; ++r) {
    const int Mrow = r + hi8;
    gLDS[Mrow * 256 + col0] = acc0[r];
    gLDS[Mrow * 256 + col1] = acc1[r];
  }
  __syncthreads();

  for (int idx = tid; idx < B_ * 64; idx += 256) {
    const int b = idx >> 6, jj = idx & 63;
    const int j = j0 + jj;
    const float gi = gLDS[b * 256 + jj]       + bih[0 * H_ + j] + bhh[0 * H_ + j];
    const float gf = gLDS[b * 256 + 64 + jj]  + bih[1 * H_ + j] + bhh[1 * H_ + j];
    const float gg = gLDS[b * 256 + 128 + jj] + bih[2 * H_ + j] + bhh[2 * H_ + j];
    const float go = gLDS[b * 256 + 192 + jj] + bih[3 * H_ + j] + bhh[3 * H_ + j];
    const float cold = c1[b * H_ + j];
    const float cn = sigf(gf) * cold + sigf(gi) * tanhf(gg);
    c1[b * H_ + j] = cn;
    h1nxt[b * H_ + j] = sigf(go) * tanhf(cn);
  }
}

// ---------------- LSTM cell 2 (M=16, N=1024, K=768) ----------------
// grid = D/64 = 4 blocks
__global__ __launch_bounds__(256) void k_lstm2(
    const float* __restrict__ h1new, const float* __restrict__ h2cur,
    const float* __restrict__ Wih, const float* __restrict__ Whh,
    const float* __restrict__ bih, const float* __restrict__ bhh,
    float* __restrict__ h2nxt, float* __restrict__ c2) {
  __shared__ float gLDS[B_ * 256];
  const int tid = threadIdx.x;
  const int wave = tid >> 5, lane = tid & 31;
  const int m = lane & 15, hi2 = (lane >> 4) * 2;
  const int j0 = blockIdx.x * 64;

  const float* arow_a = h1new + (size_t)m * H_ + hi2;  // k in [0,512)
  const float* arow_b = h2cur + (size_t)m * D_ + hi2;  // k in [512,768)

  const int tn0 = wave * 2, tn1 = tn0 + 1;
  const int gate = tn0 >> 2;
  const int n0 = gate * D_ + j0 + (tn0 & 3) * 16 + m;
  const int n1 = gate * D_ + j0 + (tn1 & 3) * 16 + m;
  const float* b0i = Wih + (size_t)n0 * 512 + hi2;
  const float* b1i = Wih + (size_t)n1 * 512 + hi2;
  const float* b0h = Whh + (size_t)n0 * 256 + hi2;
  const float* b1h = Whh + (size_t)n1 * 256 + hi2;

  v8f acc0 = {}, acc1 = {};
  gemm_seg(arow_a, b0i, b1i, 512, acc0, acc1);  // k in [0,512)
  gemm_seg(arow_b, b0h, b1h, 256, acc0, acc1);  // k in [512,768)

  const int hi8 = (hi2 >> 1) * 8;
  const int col0 = gate * 64 + (tn0 & 3) * 16 + m;
  const int col1 = gate * 64 + (tn1 & 3) * 16 + m;
#pragma unroll
  for (int r = 0; r < 8; ++r) {
    const int Mrow = r + hi8;
    gLDS[Mrow * 256 + col0] = acc0[r];
    gLDS[Mrow * 256 + col1] = acc1[r];
  }
  __syncthreads();

  for (int idx = tid; idx < B_ * 64; idx += 256) {
    const int b = idx >> 6, jj = idx & 63;
    const int j = j0 + jj;
    const float gi = gLDS[b * 256 + jj]       + bih[0 * D_ + j] + bhh[0 * D_ + j];
    const float gf = gLDS[b * 256 + 64 + jj]  + bih[1 * D_ + j] + bhh[1 * D_ + j];
    const float gg = gLDS[b * 256 + 128 + jj] + bih[2 * D_ + j] + bhh[2 * D_ + j];
    const float go = gLDS[b * 256 + 192 + jj] + bih[3 * D_ + j] + bhh[3 * D_ + j];
    const float cold = c2[b * D_ + j];
    const float cn = sigf(gf) * cold + sigf(gi) * tanhf(gg);
    c2[b * D_ + j] = cn;
    h2nxt[b * D_ + j] = sigf(go) * tanhf(cn);
  }
}

// ---------------- attention chunk: energies, exp, partial ctx ----------------
// grid = L/64 = 32 blocks; key/value are [L,B,D] (reference layout pre-transpose)
__global__ __launch_bounds__(256) void k_attn(
    const float* __restrict__ key, const float* __restrict__ value,
    const int* __restrict__ lens, const float* __restrict__ h2,
    float* __restrict__ eOut, float* __restrict__ sumpart,
    float* __restrict__ ctxpart) {
  __shared__ float sh2[B_ * 260];    // padded stride: conflict-free strided reads
  __shared__ float se[64 * 16];
  const int tid = threadIdx.x;
  const int ci = blockIdx.x;

  for (int i = tid; i < B_ * D_; i += 256) sh2[(i >> 8) * 260 + (i & 255)] = h2[i];
  __syncthreads();

  // phase 1: e[l,b] = exp(h2[b] . key[l,b]) with zero-masking
  for (int it = 0; it < 4; ++it) {
    const int p = it * 256 + tid;
    const int ll = p >> 4, b = p & 15;
    const int l = ci * 64 + ll;
    const float* kr = key + ((size_t)l * B_ + b) * D_;
    // warm WGP$ with the value row phase 2 will consume (global_prefetch path)
    __builtin_prefetch(value + ((size_t)l * B_ + b) * D_, 0, 0);
    const float* hr = sh2 + b * 260;
    float acc = 0.f;
#pragma unroll 8
    for (int d = 0; d < D_; d += 4) {
      float4 kv = *(const float4*)(kr + d);
      acc += kv.x * hr[d] + kv.y * hr[d + 1] + kv.z * hr[d + 2] + kv.w * hr[d + 3];
    }
    const float e = (l < lens[b]) ? expf(acc) : 0.0f;  // max-free; |energy| small
    se[ll * 16 + b] = e;
    eOut[(size_t)b * L_ + l] = e;
  }
  __syncthreads();

  if (tid < 16) {
    float s = 0.f;
    for (int ll = 0; ll < 64; ++ll) s += se[ll * 16 + tid];
    sumpart[ci * 16 + tid] = s;
  }

  // phase 2: ctxpart[ci,b,d] = sum_ll e[ll,b] * value[l,b,d]  (coalesced in d)
  const int d = tid;
  for (int b = 0; b < B_; ++b) {
    float acc = 0.f;
    const float* vb = value + (((size_t)ci * 64) * B_ + b) * D_ + d;
#pragma unroll 4
    for (int ll = 0; ll < 64; ++ll)
      acc += se[ll * 16 + b] * vb[(size_t)ll * B_ * D_];
    ctxpart[((size_t)ci * 16 + b) * D_ + d] = acc;
  }
}

// ---------------- finalize: Z, ctx, attn output, vocab projection ----------------
// grid = B = 16 blocks
__global__ __launch_bounds__(256) void k_fin(
    const float* __restrict__ e, const float* __restrict__ sumpart,
    const float* __restrict__ ctxpart, const float* __restrict__ h2,
    const float* __restrict__ Wout, const float* __restrict__ bout,
    float* __restrict__ ctx, float* __restrict__ outTotal,
    float* __restrict__ outAttn, int t) {
  __shared__ float sred[32];
  __shared__ float scat[2 * D_];
  __shared__ float sZ;
  const int tid = threadIdx.x;
  const int b = blockIdx.x;

  if (tid < 32) sred[tid] = sumpart[tid * 16 + b];
  __syncthreads();
  if (tid == 0) {
    float s = 0.f;
    for (int i = 0; i < 32; ++i) s += sred[i];
    sZ = s;
  }
  __syncthreads();
  const float Zinv = 1.0f / sZ;

  // ctx[b,d] = sum_ci ctxpart / Z
  {
    float a = 0.f;
#pragma unroll 8
    for (int ci = 0; ci < 32; ++ci) a += ctxpart[((size_t)ci * 16 + b) * D_ + tid];
    const float cv = a * Zinv;
    ctx[b * D_ + tid] = cv;
    scat[D_ + tid] = cv;
    scat[tid] = h2[b * D_ + tid];
  }

  // attn[t,b,l] = e / Z (masked positions have e == 0)
#pragma unroll
  for (int j = 0; j < L_ / 256; ++j) {
    const int l = j * 256 + tid;
    outAttn[((size_t)t * B_ + b) * L_ + l] = e[(size_t)b * L_ + l] * Zinv;
  }
  __syncthreads();

  // out[t,b,v] = concat(h2, ctx) . Wout[v] + bout[v]
  if (tid < V_) {
    float a = bout[tid];
    const float* wr = Wout + (size_t)tid * (2 * D_);
#pragma unroll 8
    for (int k = 0; k < 2 * D_; ++k) a += wr[k] * scat[k];
    outTotal[((size_t)t * B_ + b) * V_ + tid] = a;
  }
}

extern "C" void kernel_launch(void* const* d_in, const int* in_sizes, int n_in,
                              void* d_out, int out_size, void* d_ws, size_t ws_size,
                              hipStream_t stream) {
  const float* key   = (const float*)d_in[0];
  const float* value = (const float*)d_in[1];
  const int*   lens  = (const int*)d_in[2];
  const int*   tr    = (const int*)d_in[3];
  const float* emb   = (const float*)d_in[4];
  const float* Wih1  = (const float*)d_in[5];
  const float* Whh1  = (const float*)d_in[6];
  const float* bih1  = (const float*)d_in[7];
  const float* bhh1  = (const float*)d_in[8];
  const float* Wih2  = (const float*)d_in[9];
  const float* Whh2  = (const float*)d_in[10];
  const float* bih2  = (const float*)d_in[11];
  const float* bhh2  = (const float*)d_in[12];
  const float* Wout  = (const float*)d_in[13];
  const float* bout  = (const float*)d_in[14];
  const float* h10   = (const float*)d_in[15];
  const float* c10   = (const float*)d_in[16];
  const float* h20   = (const float*)d_in[17];
  const float* c20   = (const float*)d_in[18];

  float* W = (float*)d_ws;
  float* h1buf = W + OFF_H1;   // 2 x [16,512]
  float* c1    = W + OFF_C1;
  float* h2buf = W + OFF_H2;   // 2 x [16,256]
  float* c2    = W + OFF_C2;
  float* ctx   = W + OFF_CTX;
  float* eWS   = W + OFF_E;
  float* sumP  = W + OFF_SUM;
  float* ctxP  = W + OFF_CTXP;

  float* outTotal = (float*)d_out;                          // [T,B,V]
  float* outAttn  = (float*)d_out + (size_t)T_ * B_ * V_;   // [T,B,L]

  k_init<<<32, 256, 0, stream>>>(h10, c10, h20, c20, W);

  for (int t = 0; t < T_; ++t) {
    const int cur = t & 1, nxt = cur ^ 1;
    float* h1c = h1buf + (size_t)cur * B_ * H_;
    float* h1n = h1buf + (size_t)nxt * B_ * H_;
    float* h2c = h2buf + (size_t)cur * B_ * D_;
    float* h2n = h2buf + (size_t)nxt * B_ * D_;

    k_lstm1<<<H_ / 64, 256, 0, stream>>>(emb, tr, Wih1, Whh1, bih1, bhh1,
                                         ctx, h1c, h1n, c1, t);
    k_lstm2<<<D_ / 64, 256, 0, stream>>>(h1n, h2c, Wih2, Whh2, bih2, bhh2,
                                         h2n, c2);
    k_attn<<<L_ / 64, 256, 0, stream>>>(key, value, lens, h2n, eWS, sumP, ctxP);
    k_fin<<<B_, 256, 0, stream>>>(eWS, sumP, ctxP, h2n, Wout, bout,
                                  ctx, outTotal, outAttn, t);
  }
}